// SingleNode_Reserve_Deterministic_DA_OptNet_46926812676865
// MI455X (gfx1250) — compile-verified
//
#include <hip/hip_runtime.h>
#include <math.h>

// MI455X / gfx1250, wave32. One wave per LP for the PDHG scan (register-resident,
// structured-K closed form); V_WMMA_F32_16X16X4_F32 row-sum reduction for obj.

typedef __attribute__((ext_vector_type(2))) float v2f;
typedef __attribute__((ext_vector_type(8))) float v8f;

#define RES_UP 0.05f
#define RES_DN 0.02f
#define REQ_UP 0.02f
#define REQ_DN 0.02f

// ---------------------------------------------------------------------------
// Kernel 1: PDHG solver. One wave32 per (b,t) LP. Requires G <= 64
// (each lane owns generators g = lane and g = lane+32). All x,y state lives
// in VGPRs; the three cross-lane sums use 5-step xor butterflies.
// ---------------------------------------------------------------------------
__global__ __launch_bounds__(256) void pdhg_kernel(
    const float* __restrict__ forecast,   // (B*T)
    const float* __restrict__ pmin,       // (G)
    const float* __restrict__ pmax,       // (G)
    const float* __restrict__ bcost,      // (G)
    const int*   __restrict__ n_iters_ptr,
    float* __restrict__ Pout,             // (B,G,T)
    float* __restrict__ Ruout,            // (B,G,T)
    float* __restrict__ Rdout,            // (B,G,T)
    int G, int T, int nprob, float tau, float sigma)
{
    const int lane = threadIdx.x & 31;
    const int wave = blockIdx.x * (blockDim.x >> 5) + (threadIdx.x >> 5);
    if (wave >= nprob) return;            // wave-uniform exit

    const int   n_iters = *n_iters_ptr;
    const float D       = forecast[wave];

    // Per-lane constants for the (up to) two generators this lane owns.
    float msk[2], bc[2], bu[2], bd[2], pmx[2], pmn[2];
#pragma unroll
    for (int j = 0; j < 2; ++j) {
        const int  gi = lane + 32 * j;
        const bool ok = gi < G;
        msk[j] = ok ? 1.0f : 0.0f;
        bc[j]  = ok ? bcost[gi] : 0.0f;
        pmx[j] = ok ? pmax[gi]  : 0.0f;
        pmn[j] = ok ? pmin[gi]  : 0.0f;
        bu[j]  = RES_UP * bc[j];
        bd[j]  = RES_DN * bc[j];
    }
    const float ruq = REQ_UP * D;
    const float rdq = REQ_DN * D;

    // State: x = (P,Ru,Rd), y = (y0 | yA | yB | yC | yD | y5 | y6)
    float P[2]  = {0.f, 0.f}, Ru[2] = {0.f, 0.f}, Rd[2] = {0.f, 0.f};
    float yA[2] = {0.f, 0.f}, yB[2] = {0.f, 0.f};
    float yC[2] = {0.f, 0.f}, yD[2] = {0.f, 0.f};
    float y0 = 0.f, y5 = 0.f, y6 = 0.f;   // scalar rows, replicated per lane

    for (int it = 0; it < n_iters; ++it) {
        float Pn[2], Run[2], Rdn[2], Pb[2], Rub[2], Rdb[2];
        float sP = 0.f, sRu = 0.f, sRd = 0.f;
#pragma unroll
        for (int j = 0; j < 2; ++j) {
            // x_new = relu(x - tau*(c_obj + K^T y))
            const float gP = y0 + yA[j] - yB[j] + yC[j] - yD[j];
            Pn[j]  = fmaxf(fmaf(-tau, bc[j] + gP,            P[j]),  0.f);
            Run[j] = fmaxf(fmaf(-tau, bu[j] + yA[j] - y5,    Ru[j]), 0.f);
            Rdn[j] = fmaxf(fmaf(-tau, bd[j] + yB[j] - y6,    Rd[j]), 0.f);
            // x_bar = 2*x_new - x
            Pb[j]  = 2.f * Pn[j]  - P[j];
            Rub[j] = 2.f * Run[j] - Ru[j];
            Rdb[j] = 2.f * Rdn[j] - Rd[j];
            sP  = fmaf(msk[j], Pb[j],  sP);
            sRu = fmaf(msk[j], Rub[j], sRu);
            sRd = fmaf(msk[j], Rdb[j], sRd);
        }
        // wave32 butterfly reductions -> totals in every lane
#pragma unroll
        for (int off = 16; off >= 1; off >>= 1) {
            sP  += __shfl_xor(sP,  off, 32);
            sRu += __shfl_xor(sRu, off, 32);
            sRd += __shfl_xor(sRd, off, 32);
        }
        // y update (row 0 is the equality: no projection)
        y0 = fmaf(sigma, sP - D, y0);
        y5 = fmaxf(fmaf(sigma, ruq - sRu, y5), 0.f);
        y6 = fmaxf(fmaf(sigma, rdq - sRd, y6), 0.f);
#pragma unroll
        for (int j = 0; j < 2; ++j) {
            yA[j] = fmaxf(fmaf(sigma, Pb[j] + Rub[j] - pmx[j], yA[j]), 0.f);
            yB[j] = fmaxf(fmaf(sigma, Rdb[j] - Pb[j] + pmn[j], yB[j]), 0.f);
            yC[j] = fmaxf(fmaf(sigma, Pb[j] - pmx[j],          yC[j]), 0.f);
            yD[j] = fmaxf(fmaf(sigma, pmn[j] - Pb[j],          yD[j]), 0.f);
            P[j] = Pn[j]; Ru[j] = Run[j]; Rd[j] = Rdn[j];
        }
    }

    // Write P_DA/R_up/R_dn transposed to (B,G,T)
    const int    b    = wave / T;
    const int    t    = wave - b * T;
    const size_t base = (size_t)b * G * T + t;
#pragma unroll
    for (int j = 0; j < 2; ++j) {
        const int gi = lane + 32 * j;
        if (gi < G) {
            const size_t idx = base + (size_t)gi * T;
            Pout[idx]  = P[j];
            Ruout[idx] = Ru[j];
            Rdout[idx] = Rd[j];
        }
    }
}

// ---------------------------------------------------------------------------
// Kernel 2: epilogue. Cost_DA = b*P + c (elementwise) and obj = row-sums of the
// 128 x (G*T) contribution matrix, reduced with V_WMMA_F32_16X16X4_F32
// against B = ones. One block (8 waves) per 16 batch rows; each wave owns a
// strided set of K=4 column slices; partial D tiles combined through LDS.
// A layout (ISA 7.12.2, 32-bit A 16x4): lanes 0-15 hold K=0,1; lanes 16-31
// hold K=2,3; row M = lane&15. D: lane 0 holds M=0..7 (N=0), lane 16 M=8..15.
// ---------------------------------------------------------------------------
__global__ __launch_bounds__(256) void epilogue_kernel(
    const float* __restrict__ P,
    const float* __restrict__ Ru,
    const float* __restrict__ Rd,
    const float* __restrict__ bcost,
    const float* __restrict__ ccost,
    float* __restrict__ Cost,
    float* __restrict__ obj,
    int B, int G, int T)
{
    __shared__ float part[8][16];

    const int    lane  = threadIdx.x & 31;
    const int    wv    = threadIdx.x >> 5;        // 0..7, uniform per wave
    const int    tile  = blockIdx.x;
    const int    m     = lane & 15;
    const int    half  = lane >> 4;               // 0 -> K=0,1 ; 1 -> K=2,3
    const int    row   = tile * 16 + m;
    const int    GT    = G * T;
    const bool   rok   = row < B;
    const size_t rbase = (size_t)row * GT;
    const int    nslice = (GT + 3) >> 2;

    v2f bone; bone.x = 1.0f; bone.y = 1.0f;       // B = ones(4x16)
    v8f acc = {};

    for (int s = wv; s < nslice; s += 8) {        // uniform trip count per wave
        const int col = s * 4 + half * 2;
        float e0 = 0.f, e1 = 0.f;
        if (rok && col < GT) {
            const int   gi = col / T;
            const float bv = bcost[gi];
            const float cv = fmaf(bv, P[rbase + col], ccost[gi]);
            Cost[rbase + col] = cv;
            e0 = cv + RES_UP * bv * Ru[rbase + col] + RES_DN * bv * Rd[rbase + col];
        }
        if (rok && (col + 1) < GT) {
            const int   gi = (col + 1) / T;
            const float bv = bcost[gi];
            const float cv = fmaf(bv, P[rbase + col + 1], ccost[gi]);
            Cost[rbase + col + 1] = cv;
            e1 = cv + RES_UP * bv * Ru[rbase + col + 1] + RES_DN * bv * Rd[rbase + col + 1];
        }
        v2f a; a.x = e0; a.y = e1;
        // D(16x16) = A(16x4) * ones(4x16) + C  ==> every column of D holds the
        // running row-sums. 8 args: (neg_a, A, neg_b, B, c_mod, C, reuse_a, reuse_b)
        acc = __builtin_amdgcn_wmma_f32_16x16x4_f32(
            false, a, false, bone, (short)0, acc, false, false);
    }

    // Extract column N=0 of each wave's D tile, combine across waves in LDS.
    if (lane == 0) {
#pragma unroll
        for (int r = 0; r < 8; ++r) part[wv][r] = acc[r];        // M = 0..7
    } else if (lane == 16) {
#pragma unroll
        for (int r = 0; r < 8; ++r) part[wv][8 + r] = acc[r];    // M = 8..15
    }
    __syncthreads();
    if (threadIdx.x < 16) {
        const int orow = tile * 16 + threadIdx.x;
        if (orow < B) {
            float s = 0.f;
#pragma unroll
            for (int w = 0; w < 8; ++w) s += part[w][threadIdx.x];
            obj[orow] = s;
        }
    }
}

// ---------------------------------------------------------------------------
// Host launcher. Inputs: forecast(B*T), pmin(G), pmax(G), b(G), c(G), n_iters(1).
// Outputs flat: P_DA | R_up | R_dn | obj | Cost_DA.
// tau = sigma = 0.9/||K||_2 with the analytic spectral norm
//   ||K||_2 = sqrt((2G + 5 + sqrt(17)) / 2)
// (eigen-decomposition of K^T K into the mean direction and its complement).
// ---------------------------------------------------------------------------
extern "C" void kernel_launch(void* const* d_in, const int* in_sizes, int n_in,
                              void* d_out, int out_size, void* d_ws, size_t ws_size,
                              hipStream_t stream)
{
    const float* forecast = (const float*)d_in[0];
    const float* pmin     = (const float*)d_in[1];
    const float* pmax     = (const float*)d_in[2];
    const float* bcost    = (const float*)d_in[3];
    const float* ccost    = (const float*)d_in[4];
    const int*   n_iters  = (const int*)d_in[5];

    const long G = in_sizes[1];                 // 50
    const long S = in_sizes[0];                 // B*T = 3072
    const long B = (long)out_size - 4L * G * S; // out = B*(4*G*T + 1)
    const long T = S / B;                       // 24

    const double L     = sqrt((2.0 * (double)G + 5.0 + sqrt(17.0)) * 0.5);
    const float  tau   = (float)(0.9 / L);
    const float  sigma = tau;

    float* Pout  = (float*)d_out;
    float* Ruout = Pout  + (size_t)B * G * T;
    float* Rdout = Ruout + (size_t)B * G * T;
    float* objp  = Rdout + (size_t)B * G * T;
    float* Cost  = objp  + (size_t)B;

    const int wavesPerBlock = 8;
    const int grid1 = (int)((S + wavesPerBlock - 1) / wavesPerBlock);
    pdhg_kernel<<<grid1, 256, 0, stream>>>(forecast, pmin, pmax, bcost, n_iters,
                                           Pout, Ruout, Rdout,
                                           (int)G, (int)T, (int)S, tau, sigma);

    const int ntiles = (int)((B + 15) / 16);
    epilogue_kernel<<<ntiles, 256, 0, stream>>>(Pout, Ruout, Rdout, bcost, ccost,
                                                Cost, objp, (int)B, (int)G, (int)T);
}